// GraphConv_54778012893227
// MI455X (gfx1250) — compile-verified
//
#include <hip/hip_runtime.h>

typedef float v2f __attribute__((ext_vector_type(2)));
typedef float v8f __attribute__((ext_vector_type(8)));

#define N_NODES 100000
#define D_IN    128
#define D_OUT   512

// ---------------------------------------------------------------------------
// Phase 1: agg[col] += w * x[row]. One wave32 per edge; each lane owns 4
// consecutive floats (float4 load of x row, 4 f32 global atomics into agg).
// x (51MB) and agg (51MB) both fit the 192MB L2, so atomics resolve in L2.
// ---------------------------------------------------------------------------
__global__ __launch_bounds__(256) void graphconv_scatter(
    const float* __restrict__ x,
    const long long* __restrict__ edge_index,   // [2, E] int64
    const float* __restrict__ edge_weight,      // [E]
    float* __restrict__ agg,                    // [N, 128] (pre-zeroed)
    int E)
{
    const int wave = blockIdx.x * (blockDim.x >> 5) + (threadIdx.x >> 5);
    if (wave >= E) return;
    const int lane = threadIdx.x & 31;

    const long long row = edge_index[wave];                 // source node
    const long long col = edge_index[(long long)E + wave];  // destination node
    const float     w   = edge_weight[wave];

    const float4 v = ((const float4*)(x + (size_t)row * D_IN))[lane];
    float* dst = agg + (size_t)col * D_IN + lane * 4;
    atomicAdd(dst + 0, w * v.x);
    atomicAdd(dst + 1, w * v.y);
    atomicAdd(dst + 2, w * v.z);
    atomicAdd(dst + 3, w * v.w);
}

// ---------------------------------------------------------------------------
// Phase 2: out = agg @ Wl^T + x @ Wr^T + bl using V_WMMA_F32_16X16X4_F32.
// One wave computes a 64x16 output strip = four 16x16 C tiles; B fragments
// (Wl/Wr) are loaded once per K-step and reused by 4 WMMAs each. Both GEMMs
// accumulate into the same C tiles (fuses the "+").
//
// f32 WMMA fragment layout (wave32):
//   A 16x4 : lane L holds A[M = L&15][K = k0 + 2*(L>>4) + {0,1}]  -> float2
//   B 4x16 : lane L holds B[K = k0 + 2*(L>>4) + {0,1}][N = L&15]  -> float2
//            B[k][n] = W[n][k] (W is [D_OUT, D_IN] row-major)
//   C 16x16: VGPR j, lane L -> row (j + 8*(L>>4)), col (L&15)
// ---------------------------------------------------------------------------
__global__ __launch_bounds__(256) void graphconv_gemm(
    const float* __restrict__ agg,   // [N, 128]
    const float* __restrict__ x,     // [N, 128]
    const float* __restrict__ Wl,    // [512, 128]
    const float* __restrict__ Wr,    // [512, 128]
    const float* __restrict__ bl,    // [512]
    float* __restrict__ out)         // [N, 512]
{
    const int lane   = threadIdx.x & 31;
    const int waveId = threadIdx.x >> 5;          // 0..7
    const int hi     = lane >> 4;                 // 0 or 1
    const int l15    = lane & 15;

    const int mbase = blockIdx.x * 64;                  // 4 sub-tiles of 16 rows
    const int nbase = blockIdx.y * 128 + waveId * 16;   // 16 output features

    // Row pointers for the 4 A sub-tiles (clamped: safe reads, stores guarded)
    int r0 = mbase +  0 + l15; if (r0 >= N_NODES) r0 = N_NODES - 1;
    int r1 = mbase + 16 + l15; if (r1 >= N_NODES) r1 = N_NODES - 1;
    int r2 = mbase + 32 + l15; if (r2 >= N_NODES) r2 = N_NODES - 1;
    int r3 = mbase + 48 + l15; if (r3 >= N_NODES) r3 = N_NODES - 1;

    const float* aggR0 = agg + (size_t)r0 * D_IN;
    const float* aggR1 = agg + (size_t)r1 * D_IN;
    const float* aggR2 = agg + (size_t)r2 * D_IN;
    const float* aggR3 = agg + (size_t)r3 * D_IN;
    const float* xR0   = x   + (size_t)r0 * D_IN;
    const float* xR1   = x   + (size_t)r1 * D_IN;
    const float* xR2   = x   + (size_t)r2 * D_IN;
    const float* xR3   = x   + (size_t)r3 * D_IN;

    const float* wlB = Wl + (size_t)(nbase + l15) * D_IN;  // column n of B = row n of W
    const float* wrB = Wr + (size_t)(nbase + l15) * D_IN;

    v8f c0 = {}, c1 = {}, c2 = {}, c3 = {};

    #pragma unroll 4
    for (int k0 = 0; k0 < D_IN; k0 += 4) {
        const int ko = k0 + 2 * hi;
        const v2f bL = *(const v2f*)(wlB + ko);   // Wl^T fragment
        const v2f bR = *(const v2f*)(wrB + ko);   // Wr^T fragment

        v2f a;
        a = *(const v2f*)(aggR0 + ko);
        c0 = __builtin_amdgcn_wmma_f32_16x16x4_f32(false, a, false, bL, (short)0, c0, false, false);
        a = *(const v2f*)(xR0 + ko);
        c0 = __builtin_amdgcn_wmma_f32_16x16x4_f32(false, a, false, bR, (short)0, c0, false, false);

        a = *(const v2f*)(aggR1 + ko);
        c1 = __builtin_amdgcn_wmma_f32_16x16x4_f32(false, a, false, bL, (short)0, c1, false, false);
        a = *(const v2f*)(xR1 + ko);
        c1 = __builtin_amdgcn_wmma_f32_16x16x4_f32(false, a, false, bR, (short)0, c1, false, false);

        a = *(const v2f*)(aggR2 + ko);
        c2 = __builtin_amdgcn_wmma_f32_16x16x4_f32(false, a, false, bL, (short)0, c2, false, false);
        a = *(const v2f*)(xR2 + ko);
        c2 = __builtin_amdgcn_wmma_f32_16x16x4_f32(false, a, false, bR, (short)0, c2, false, false);

        a = *(const v2f*)(aggR3 + ko);
        c3 = __builtin_amdgcn_wmma_f32_16x16x4_f32(false, a, false, bL, (short)0, c3, false, false);
        a = *(const v2f*)(xR3 + ko);
        c3 = __builtin_amdgcn_wmma_f32_16x16x4_f32(false, a, false, bR, (short)0, c3, false, false);
    }

    const float bias = bl[nbase + l15];
    const size_t colOff = (size_t)nbase + l15;

    // Store C tiles: VGPR j -> row (tilebase + j + 8*hi), col l15
    #define STORE_TILE(C, TB)                                                  \
        if ((TB) < N_NODES) {                                                  \
            const int rowb = (TB) + 8 * hi;                                    \
            _Pragma("unroll")                                                  \
            for (int j = 0; j < 8; ++j)                                        \
                out[(size_t)(rowb + j) * D_OUT + colOff] = (C)[j] + bias;      \
        }
    STORE_TILE(c0, mbase +  0)
    STORE_TILE(c1, mbase + 16)
    STORE_TILE(c2, mbase + 32)
    STORE_TILE(c3, mbase + 48)
    #undef STORE_TILE
}

extern "C" void kernel_launch(void* const* d_in, const int* in_sizes, int n_in,
                              void* d_out, int out_size, void* d_ws, size_t ws_size,
                              hipStream_t stream) {
    const float*     x           = (const float*)d_in[0];
    const long long* edge_index  = (const long long*)d_in[1];  // int64 [2,E]
    const float*     edge_weight = (const float*)d_in[2];
    const float*     Wl          = (const float*)d_in[3];
    const float*     bl          = (const float*)d_in[4];
    const float*     Wr          = (const float*)d_in[5];
    float*           out         = (float*)d_out;
    float*           agg         = (float*)d_ws;               // [N,128] = 51.2 MB

    const int E = in_sizes[2];  // edge_weight element count

    // Zero the aggregation buffer (graph-capture legal).
    hipMemsetAsync(agg, 0, (size_t)N_NODES * D_IN * sizeof(float), stream);

    // Phase 1: edge scatter. 8 waves per 256-thread block, one wave per edge.
    const int wavesPerBlock = 256 / 32;
    const int sblocks = (E + wavesPerBlock - 1) / wavesPerBlock;
    graphconv_scatter<<<sblocks, 256, 0, stream>>>(x, edge_index, edge_weight, agg, E);

    // Phase 2: fused dual GEMM + bias via f32 WMMA.
    dim3 grid((N_NODES + 63) / 64, D_OUT / 128);
    graphconv_gemm<<<grid, 256, 0, stream>>>(agg, x, Wl, Wr, bl, out);
}